// MaskedMGN_44985487458935
// MI455X (gfx1250) — compile-verified
//
#include <hip/hip_runtime.h>
#include <cstdint>

#define NN 100000
#define NE 600000
#define HD 128
#define CO 3
#define LDH 132
#define LN_EPS 1e-5f

typedef __attribute__((ext_vector_type(2))) float v2f;
typedef __attribute__((ext_vector_type(8))) float v8f;

__device__ __forceinline__ v8f wmma_f32(v2f a, v2f b, v8f c) {
    // V_WMMA_F32_16X16X4_F32: (neg_a, A, neg_b, B, c_mod, C, reuse_a, reuse_b)
    return __builtin_amdgcn_wmma_f32_16x16x4_f32(false, a, false, b, (short)0, c, false, false);
}

// One 16x16 output tile per wave. A (16xK) comes from LDS (row-major, stride ldA),
// B fragments are read directly from the global weight matrix W[K][128].
// A layout (ISA 16x4 f32): lanes 0-15 hold (K0,K1) of row M=lane; lanes 16-31 hold (K2,K3).
// B layout (C/D analogy): VGPR0 = rows {k0 | k0+2}, VGPR1 = rows {k0+1 | k0+3}, N = lane&15.
__device__ __forceinline__ v8f gemm_tile(const float* Asub, int ldA, int K, int K4,
                                         const float* __restrict__ W) {
    const int lane = threadIdx.x & 31;
    const int wid  = threadIdx.x >> 5;
    const int half2 = (lane >> 4) << 1;
    const int rowA = lane & 15;
    const int col  = wid * 16 + rowA;
    const float* arow = Asub + rowA * ldA;
    v8f acc = {0.f, 0.f, 0.f, 0.f, 0.f, 0.f, 0.f, 0.f};
    const int Kmain = K & ~3;
    for (int k0 = 0; k0 < Kmain; k0 += 4) {
        const int koff = k0 + half2;
        v2f a = *(const v2f*)(arow + koff);
        v2f b;
        b.x = W[koff * HD + col];
        b.y = W[koff * HD + HD + col];
        acc = wmma_f32(a, b, acc);
    }
    if (Kmain < K4) {  // zero-padded tail (only when K not a multiple of 4)
        const int koff = Kmain + half2;
        v2f a = *(const v2f*)(arow + koff);
        v2f b;
        b.x = (koff     < K) ? W[koff * HD + col] : 0.f;
        b.y = (koff + 1 < K) ? W[koff * HD + HD + col] : 0.f;
        acc = wmma_f32(a, b, acc);
    }
    return acc;
}

// Full 2-layer MLP on a 16-row tile: Y = relu(A@W1+b1)@W2+b2 left in Ysub (LDS).
__device__ __forceinline__ void mlp_pair(const float* Asub, int ldA, int K, int K4,
                                         float* Hsub, float* Ysub,
                                         const float* __restrict__ W1, const float* __restrict__ b1,
                                         const float* __restrict__ W2, const float* __restrict__ b2) {
    const int lane = threadIdx.x & 31;
    const int wid  = threadIdx.x >> 5;
    const int half = lane >> 4;
    const int col  = wid * 16 + (lane & 15);
    v8f h = gemm_tile(Asub, ldA, K, K4, W1);
    const float bb = b1[col];
#pragma unroll
    for (int v = 0; v < 8; ++v)
        Hsub[(v + half * 8) * LDH + col] = fmaxf(h[v] + bb, 0.f);
    __syncthreads();
    v8f y = gemm_tile(Hsub, LDH, HD, HD, W2);
    const float b2v = b2[col];
#pragma unroll
    for (int v = 0; v < 8; ++v)
        Ysub[(v + half * 8) * LDH + col] = y[v] + b2v;
    __syncthreads();
}

// Row-wise LayerNorm statistics: 16 threads per row, shfl_xor tree over width 16.
__device__ __forceinline__ void ln_stats(const float* Ysub, float vals[8], float& mu, float& rstd) {
    const int r = threadIdx.x >> 4, sub = threadIdx.x & 15;
    const float* yr = Ysub + r * LDH + sub * 8;
    float s = 0.f, s2 = 0.f;
#pragma unroll
    for (int i = 0; i < 8; ++i) { float v = yr[i]; vals[i] = v; s += v; s2 += v * v; }
#pragma unroll
    for (int o = 8; o >= 1; o >>= 1) {
        s  += __shfl_xor(s,  o, 16);
        s2 += __shfl_xor(s2, o, 16);
    }
    mu = s * (1.f / HD);
    float var = s2 * (1.f / HD) - mu * mu;
    rstd = rsqrtf(var + LN_EPS);
}

__global__ void zero_kernel(float* __restrict__ p, int n) {
    for (int i = blockIdx.x * blockDim.x + threadIdx.x; i < n; i += gridDim.x * blockDim.x)
        p[i] = 0.f;
}

// Encoder: out = LN(MLP(x)), x is [M,K] dense (K = 7 or 4).
__global__ __launch_bounds__(256) void enc_kernel(
        const float* __restrict__ x, int M, int K, int K4, int ldA,
        const float* __restrict__ W1, const float* __restrict__ b1,
        const float* __restrict__ W2, const float* __restrict__ b2,
        const float* __restrict__ g, const float* __restrict__ be,
        float* __restrict__ out) {
    __shared__ __align__(16) float Asub[16 * 12];
    __shared__ __align__(16) float Hsub[16 * LDH];
    __shared__ __align__(16) float Ysub[16 * LDH];
    const int tile = blockIdx.x;
    const int t = threadIdx.x;
    if (t < 16 * ldA) {
        int r = t / ldA, c = t % ldA;
        int row = tile * 16 + r;
        float v = 0.f;
        if (row < M && c < K) v = x[(size_t)row * K + c];
        Asub[r * ldA + c] = v;
    }
    __syncthreads();
    mlp_pair(Asub, ldA, K, K4, Hsub, Ysub, W1, b1, W2, b2);
    float vals[8], mu, rstd;
    ln_stats(Ysub, vals, mu, rstd);
    const int r = t >> 4, sub = t & 15;
    const int row = tile * 16 + r;
    if (row < M) {
        float* orow = out + (size_t)row * HD + sub * 8;
#pragma unroll
        for (int i = 0; i < 8; ++i) {
            int c = sub * 8 + i;
            orow[i] = (vals[i] - mu) * rstd * g[c] + be[c];
        }
    }
}

// Edge update: he[e] += LN(MLP(cat[hn[dst], hn[src], he[e]])) (K=384),
// fused masked scatter: msg[src[e]] += he_new[e] * emask.
__global__ __launch_bounds__(256) void edge_kernel(
        const float* __restrict__ hn, float* he,
        const int* __restrict__ srcI, const int* __restrict__ dstI,
        const int* __restrict__ maskI,
        const float* __restrict__ W1, const float* __restrict__ b1,
        const float* __restrict__ W2, const float* __restrict__ b2,
        const float* __restrict__ g, const float* __restrict__ be,
        float* msg) {
    __shared__ __align__(16) float Asub[16 * 388];
    __shared__ __align__(16) float Hsub[16 * LDH];
    __shared__ __align__(16) float Ysub[16 * LDH];
    __shared__ int s_src[16], s_dst[16], s_m[16];
    const int tile = blockIdx.x;
    const int t = threadIdx.x;
    if (t < 16) {
        int e = tile * 16 + t;
        int s = 0, d = 0, m = 0;
        if (e < NE) {
            s = srcI[e]; d = dstI[e];
            m = (maskI[s] != 0) & (maskI[d] != 0);
        }
        s_src[t] = s; s_dst[t] = d; s_m[t] = m;
    }
    __syncthreads();
    const int r = t >> 4, sub = t & 15;
    const int e = tile * 16 + r;
    const bool valid = e < NE;
    {
        const float* p0 = hn + (size_t)s_dst[r] * HD;
        const float* p1 = hn + (size_t)s_src[r] * HD;
        const float* p2 = he + (size_t)e * HD;
#pragma unroll
        for (int j = 0; j < 6; ++j) {
            int c = (sub + j * 16) * 4;
            float4 v = make_float4(0.f, 0.f, 0.f, 0.f);
            if (valid) {
                const float* p = (c < 128) ? (p0 + c) : (c < 256) ? (p1 + c - 128) : (p2 + c - 256);
                v = *(const float4*)p;
            }
            *(float4*)(Asub + r * 388 + c) = v;
        }
    }
    __syncthreads();
    mlp_pair(Asub, 388, 384, 384, Hsub, Ysub, W1, b1, W2, b2);
    float vals[8], mu, rstd;
    ln_stats(Ysub, vals, mu, rstd);
    if (valid) {
        float* hrow = he + (size_t)e * HD + sub * 8;
        float outv[8];
#pragma unroll
        for (int i = 0; i < 8; ++i) {
            int c = sub * 8 + i;
            float y = (vals[i] - mu) * rstd * g[c] + be[c];
            outv[i] = hrow[i] + y;
            hrow[i] = outv[i];
        }
        if (s_m[r]) {
            float* mrow = msg + (size_t)s_src[r] * HD + sub * 8;
#pragma unroll
            for (int i = 0; i < 8; ++i)
                unsafeAtomicAdd(mrow + i, outv[i]);
        }
    }
}

// Node update: hn[i] += LN(MLP(cat[msg[i], hn[i]])) (K=256).
__global__ __launch_bounds__(256) void node_kernel(
        const float* __restrict__ msg, float* hn,
        const float* __restrict__ W1, const float* __restrict__ b1,
        const float* __restrict__ W2, const float* __restrict__ b2,
        const float* __restrict__ g, const float* __restrict__ be) {
    __shared__ __align__(16) float Asub[16 * 260];
    __shared__ __align__(16) float Hsub[16 * LDH];
    __shared__ __align__(16) float Ysub[16 * LDH];
    const int tile = blockIdx.x;
    const int t = threadIdx.x;
    const int r = t >> 4, sub = t & 15;
    const int row = tile * 16 + r;
    const bool valid = row < NN;
    {
        const float* p0 = msg + (size_t)row * HD;
        const float* p1 = hn + (size_t)row * HD;
#pragma unroll
        for (int j = 0; j < 4; ++j) {
            int c = (sub + j * 16) * 4;
            float4 v = make_float4(0.f, 0.f, 0.f, 0.f);
            if (valid) {
                const float* p = (c < 128) ? (p0 + c) : (p1 + c - 128);
                v = *(const float4*)p;
            }
            *(float4*)(Asub + r * 260 + c) = v;
        }
    }
    __syncthreads();
    mlp_pair(Asub, 260, 256, 256, Hsub, Ysub, W1, b1, W2, b2);
    float vals[8], mu, rstd;
    ln_stats(Ysub, vals, mu, rstd);
    if (valid) {
        float* hrow = hn + (size_t)row * HD + sub * 8;
#pragma unroll
        for (int i = 0; i < 8; ++i) {
            int c = sub * 8 + i;
            hrow[i] = hrow[i] + (vals[i] - mu) * rstd * g[c] + be[c];
        }
    }
}

// Decoder: out = (relu(hn@W1+b1)@W2+b2) * mask, W2 is 128x3.
__global__ __launch_bounds__(256) void dec_kernel(
        const float* __restrict__ hn,
        const float* __restrict__ W1, const float* __restrict__ b1,
        const float* __restrict__ W2, const float* __restrict__ b2,
        const int* __restrict__ maskI,
        float* __restrict__ out) {
    __shared__ __align__(16) float Asub[16 * 132];
    __shared__ __align__(16) float Hsub[16 * LDH];
    const int tile = blockIdx.x;
    const int t = threadIdx.x;
    const int r = t >> 4, sub = t & 15;
    const int row = tile * 16 + r;
    const bool valid = row < NN;
    {
        const float* p0 = hn + (size_t)row * HD;
#pragma unroll
        for (int j = 0; j < 2; ++j) {
            int c = (sub + j * 16) * 4;
            float4 v = make_float4(0.f, 0.f, 0.f, 0.f);
            if (valid) v = *(const float4*)(p0 + c);
            *(float4*)(Asub + r * 132 + c) = v;
        }
    }
    __syncthreads();
    {
        const int lane = t & 31;
        const int wid  = t >> 5;
        const int half = lane >> 4;
        const int col  = wid * 16 + (lane & 15);
        v8f h = gemm_tile(Asub, 132, HD, HD, W1);
        const float bb = b1[col];
#pragma unroll
        for (int v = 0; v < 8; ++v)
            Hsub[(v + half * 8) * LDH + col] = fmaxf(h[v] + bb, 0.f);
    }
    __syncthreads();
    if (t < 16 * CO) {
        const int rr = t / CO, c = t % CO;
        float sum = b2[c];
        for (int k = 0; k < HD; ++k)
            sum += Hsub[rr * LDH + k] * W2[k * CO + c];
        const int orow = tile * 16 + rr;
        if (orow < NN)
            out[(size_t)orow * CO + c] = sum * ((maskI[orow] != 0) ? 1.f : 0.f);
    }
}

extern "C" void kernel_launch(void* const* d_in, const int* in_sizes, int n_in,
                              void* d_out, int out_size, void* d_ws, size_t ws_size,
                              hipStream_t stream) {
    struct MLP { const float *W1, *b1, *W2, *b2, *g, *be; };
    const float *xn = nullptr, *xe = nullptr;
    const int *ei = nullptr, *mask = nullptr;
    MLP encN{}, encE{}, nodeL[3]{}, edgeL[3]{};
    const float *decW1 = nullptr, *decb1 = nullptr, *decW2 = nullptr, *decb2 = nullptr;
    auto F = [&](int i) { return (const float*)d_in[i]; };
    if (in_sizes[0] == NN * 7) {
        // Dict-insertion flattening: xn, xe, edge_index, mask, params{enc_n,enc_e,layers,dec}
        xn = F(0); xe = F(1); ei = (const int*)d_in[2]; mask = (const int*)d_in[3];
        auto mlp6 = [&](int b, MLP& m) { m.W1 = F(b); m.b1 = F(b+1); m.W2 = F(b+2);
                                         m.b2 = F(b+3); m.g = F(b+4); m.be = F(b+5); };
        mlp6(4, encN); mlp6(10, encE);
        for (int i = 0; i < 3; ++i) { mlp6(16 + 12*i, nodeL[i]); mlp6(16 + 12*i + 6, edgeL[i]); }
        decW1 = F(52); decb1 = F(53); decW2 = F(54); decb2 = F(55);
    } else {
        // jax tree_flatten (sorted dict keys): edge_index, mask, params{dec,enc_e,enc_n,layers}, xe, xn
        ei = (const int*)d_in[0]; mask = (const int*)d_in[1];
        decW1 = F(2); decW2 = F(3); decb1 = F(4); decb2 = F(5);
        auto mlp6 = [&](int b, MLP& m) { m.W1 = F(b); m.W2 = F(b+1); m.b1 = F(b+2);
                                         m.b2 = F(b+3); m.be = F(b+4); m.g = F(b+5); };
        mlp6(6, encE); mlp6(12, encN);
        for (int i = 0; i < 3; ++i) { mlp6(18 + 12*i, edgeL[i]); mlp6(18 + 12*i + 6, nodeL[i]); }
        xe = F(54); xn = F(55);
    }
    float* hn  = (float*)d_ws;
    float* he  = hn + (size_t)NN * HD;
    float* msg = he + (size_t)NE * HD;
    const int* srcI = ei;         // edge_index[0]
    const int* dstI = ei + NE;    // edge_index[1]
    const int gridN = (NN + 15) / 16;
    const int gridE = (NE + 15) / 16;

    enc_kernel<<<gridN, 256, 0, stream>>>(xn, NN, 7, 8, 12,
        encN.W1, encN.b1, encN.W2, encN.b2, encN.g, encN.be, hn);
    enc_kernel<<<gridE, 256, 0, stream>>>(xe, NE, 4, 4, 8,
        encE.W1, encE.b1, encE.W2, encE.b2, encE.g, encE.be, he);
    for (int l = 0; l < 3; ++l) {
        zero_kernel<<<2048, 256, 0, stream>>>(msg, NN * HD);
        edge_kernel<<<gridE, 256, 0, stream>>>(hn, he, srcI, dstI, mask,
            edgeL[l].W1, edgeL[l].b1, edgeL[l].W2, edgeL[l].b2, edgeL[l].g, edgeL[l].be, msg);
        node_kernel<<<gridN, 256, 0, stream>>>(msg, hn,
            nodeL[l].W1, nodeL[l].b1, nodeL[l].W2, nodeL[l].b2, nodeL[l].g, nodeL[l].be);
    }
    dec_kernel<<<gridN, 256, 0, stream>>>(hn, decW1, decb1, decW2, decb2, mask, (float*)d_out);
    (void)n_in; (void)out_size; (void)ws_size;
}